// OurGAT_75273596830286
// MI455X (gfx1250) — compile-verified
//
#include <hip/hip_runtime.h>
#include <math.h>

typedef float v2f __attribute__((ext_vector_type(2)));
typedef float v8f __attribute__((ext_vector_type(8)));

constexpr int N_NODES = 50000;
constexpr int F_IN    = 256;
constexpr int HEADS   = 8;
constexpr int C1      = 16;
constexpr int HID     = HEADS * C1;   // 128
constexpr int NCLS    = 40;
constexpr int NCLSP   = 48;           // padded to 3x16 for WMMA
constexpr int E_EDGES = 800000;
constexpr int ETOT    = E_EDGES + N_NODES;  // 850000
constexpr float NEG_SLOPE = 0.2f;

// ---------------- helpers ----------------
__device__ __forceinline__ void atomicMaxF(float* addr, float v) {
    // sign-aware float max via integer atomics (works with -inf init)
    if (v >= 0.0f) atomicMax((int*)addr, __float_as_int(v));
    else           atomicMin((unsigned int*)addr, __float_as_uint(v));
}

// ---------------- GEMM1: h1 = x @ W1  (fp32 WMMA 16x16x4) ----------------
// block = 256 thr = 8 waves; each block: 16 rows, wave w -> 16-col tile
__global__ __launch_bounds__(256) void gemm1_wmma(
    const float* __restrict__ x, const float* __restrict__ W1, float* __restrict__ h1)
{
    __shared__ float lx[16 * F_IN]; // 16KB A tile shared by all 8 waves
    const int row0 = blockIdx.x * 16;
    for (int i = threadIdx.x; i < 16 * F_IN; i += 256) {
        int r = i >> 8, c = i & (F_IN - 1);
        lx[i] = x[(size_t)(row0 + r) * F_IN + c];
    }
    __syncthreads();

    const int wave = threadIdx.x >> 5;      // 0..7 -> column tile
    const int lane = threadIdx.x & 31;
    const int hi   = lane >> 4;             // lane-half selects K pair
    const int lo   = lane & 15;
    const int n0   = wave * 16;

    v8f c = {};
    for (int k0 = 0; k0 < F_IN; k0 += 4) {
        const int ka = k0 + 2 * hi;
        v2f a, b;
        a.x = lx[lo * F_IN + ka];
        a.y = lx[lo * F_IN + ka + 1];
        b.x = W1[(size_t)ka * HID + n0 + lo];
        b.y = W1[(size_t)(ka + 1) * HID + n0 + lo];
        c = __builtin_amdgcn_wmma_f32_16x16x4_f32(false, a, false, b, (short)0, c, false, false);
    }
#pragma unroll
    for (int r = 0; r < 8; ++r) {
        const int m = r + 8 * hi; // C/D layout: VGPR r, lane-half -> M
        h1[(size_t)(row0 + m) * HID + n0 + lo] = c[r];
    }
}

// ---------------- layer-1 per-node attention coeffs + init ----------------
__global__ void attn1_node(const float* __restrict__ h1,
                           const float* __restrict__ att_src, const float* __restrict__ att_dst,
                           float* aS, float* aD, float* m, float* dn, float* out1)
{
    int idx = blockIdx.x * blockDim.x + threadIdx.x;
    if (idx >= N_NODES * HEADS) return;
    const int n = idx / HEADS, hd = idx % HEADS;
    const float* hp = h1 + (size_t)n * HID + hd * C1;
    float s = 0.f, d = 0.f;
#pragma unroll
    for (int c = 0; c < C1; ++c) {
        float v = hp[c];
        s += v * att_src[hd * C1 + c];
        d += v * att_dst[hd * C1 + c];
    }
    aS[idx] = s; aD[idx] = d;
    m[idx]  = -INFINITY; dn[idx] = 0.f;
    float* o = out1 + (size_t)n * HID + hd * C1;
#pragma unroll
    for (int c = 0; c < C1; ++c) o[c] = 0.f;
}

// ---------------- layer-1 edge passes ----------------
__global__ void edge1_max(const int* __restrict__ src, const int* __restrict__ dst,
                          const float* __restrict__ aS, const float* __restrict__ aD, float* m)
{
    int e = blockIdx.x * blockDim.x + threadIdx.x;
    if (e >= ETOT) return;
    int s = (e < E_EDGES) ? src[e] : (e - E_EDGES);
    int d = (e < E_EDGES) ? dst[e] : (e - E_EDGES);
#pragma unroll
    for (int h = 0; h < HEADS; ++h) {
        float v = aS[s * HEADS + h] + aD[d * HEADS + h];
        v = (v > 0.f) ? v : NEG_SLOPE * v;
        atomicMaxF(&m[d * HEADS + h], v);
    }
}

__global__ void edge1_sum(const int* __restrict__ src, const int* __restrict__ dst,
                          const float* __restrict__ aS, const float* __restrict__ aD,
                          const float* __restrict__ m, float* __restrict__ p1, float* dn)
{
    int e = blockIdx.x * blockDim.x + threadIdx.x;
    if (e >= ETOT) return;
    int s = (e < E_EDGES) ? src[e] : (e - E_EDGES);
    int d = (e < E_EDGES) ? dst[e] : (e - E_EDGES);
#pragma unroll
    for (int h = 0; h < HEADS; ++h) {
        float v = aS[s * HEADS + h] + aD[d * HEADS + h];
        v = (v > 0.f) ? v : NEG_SLOPE * v;
        float mm = m[d * HEADS + h];
        if (!(mm > -INFINITY)) mm = 0.f;     // where(isfinite(m), m, 0)
        float p = expf(v - mm);
        p1[(size_t)e * HEADS + h] = p;
        atomicAdd(&dn[d * HEADS + h], p);
    }
}

// block = 128 thr (one per channel), EPB1 edges per block
constexpr int EPB1 = 8;
__global__ __launch_bounds__(128) void edge1_agg(
    const int* __restrict__ src, const int* __restrict__ dst,
    const float* __restrict__ p1, const float* __restrict__ dn,
    const float* __restrict__ h1, float* __restrict__ out1)
{
    const int t = threadIdx.x;      // channel 0..127
    const int head = t >> 4;
    const int e0 = blockIdx.x * EPB1;
#pragma unroll
    for (int i = 0; i < EPB1; ++i) {
        const int e = e0 + i;
        if (e >= ETOT) break;
        int s = (e < E_EDGES) ? src[e] : (e - E_EDGES);
        int d = (e < E_EDGES) ? dst[e] : (e - E_EDGES);
        float alpha = p1[(size_t)e * HEADS + head] / (dn[d * HEADS + head] + 1e-16f);
        atomicAdd(&out1[(size_t)d * HID + t], alpha * h1[(size_t)s * HID + t]);
    }
}

// ---------------- mid: h = elu(out1 + b1), written in-place over h1 ----------------
__global__ void node_mid(const float* __restrict__ out1, const float* __restrict__ b1,
                         float* __restrict__ h)
{
    int idx = blockIdx.x * blockDim.x + threadIdx.x;
    if (idx >= N_NODES * HID) return;
    float v = out1[idx] + b1[idx & (HID - 1)];
    h[idx] = (v > 0.f) ? v : expm1f(v);
}

// ---------------- GEMM2: h2 = h @ W2 (fp32 WMMA, 40 cols padded to 48) ----------------
// block = 96 thr = 3 waves; 16 rows per block
__global__ __launch_bounds__(96) void gemm2_wmma(
    const float* __restrict__ h, const float* __restrict__ W2, float* __restrict__ h2)
{
    __shared__ float lh[16 * HID];     // 8KB
    __shared__ float lw[HID * NCLSP];  // 24KB, zero-padded cols 40..47
    const int row0 = blockIdx.x * 16;
    for (int i = threadIdx.x; i < 16 * HID; i += 96)
        lh[i] = h[(size_t)(row0 + (i >> 7)) * HID + (i & (HID - 1))];
    for (int i = threadIdx.x; i < HID * NCLSP; i += 96) {
        int k = i / NCLSP, cc = i % NCLSP;
        lw[i] = (cc < NCLS) ? W2[k * NCLS + cc] : 0.0f;
    }
    __syncthreads();

    const int wave = threadIdx.x >> 5;   // 0..2
    const int lane = threadIdx.x & 31;
    const int hi = lane >> 4, lo = lane & 15;
    const int n0 = wave * 16;

    v8f c = {};
    for (int k0 = 0; k0 < HID; k0 += 4) {
        const int ka = k0 + 2 * hi;
        v2f a, b;
        a.x = lh[lo * HID + ka];
        a.y = lh[lo * HID + ka + 1];
        b.x = lw[ka * NCLSP + n0 + lo];
        b.y = lw[(ka + 1) * NCLSP + n0 + lo];
        c = __builtin_amdgcn_wmma_f32_16x16x4_f32(false, a, false, b, (short)0, c, false, false);
    }
#pragma unroll
    for (int r = 0; r < 8; ++r) {
        const int m = r + 8 * hi;
        const int col = n0 + lo;
        if (col < NCLS) h2[(size_t)(row0 + m) * NCLS + col] = c[r];
    }
}

// ---------------- layer-2 node coeffs + init ----------------
__global__ void attn2_node(const float* __restrict__ h2,
                           const float* __restrict__ att_src2, const float* __restrict__ att_dst2,
                           float* aS, float* aD, float* m, float* dn, float* out2)
{
    int n = blockIdx.x * blockDim.x + threadIdx.x;
    if (n >= N_NODES) return;
    float s = 0.f, d = 0.f;
#pragma unroll
    for (int c = 0; c < NCLS; ++c) {
        float v = h2[(size_t)n * NCLS + c];
        s += v * att_src2[c];
        d += v * att_dst2[c];
        out2[(size_t)n * NCLS + c] = 0.f;
    }
    aS[n] = s; aD[n] = d; m[n] = -INFINITY; dn[n] = 0.f;
}

__global__ void edge2_max(const int* __restrict__ src, const int* __restrict__ dst,
                          const float* __restrict__ aS, const float* __restrict__ aD, float* m)
{
    int e = blockIdx.x * blockDim.x + threadIdx.x;
    if (e >= ETOT) return;
    int s = (e < E_EDGES) ? src[e] : (e - E_EDGES);
    int d = (e < E_EDGES) ? dst[e] : (e - E_EDGES);
    float v = aS[s] + aD[d];
    v = (v > 0.f) ? v : NEG_SLOPE * v;
    atomicMaxF(&m[d], v);
}

__global__ void edge2_sum(const int* __restrict__ src, const int* __restrict__ dst,
                          const float* __restrict__ aS, const float* __restrict__ aD,
                          const float* __restrict__ m, float* __restrict__ p2, float* dn)
{
    int e = blockIdx.x * blockDim.x + threadIdx.x;
    if (e >= ETOT) return;
    int s = (e < E_EDGES) ? src[e] : (e - E_EDGES);
    int d = (e < E_EDGES) ? dst[e] : (e - E_EDGES);
    float v = aS[s] + aD[d];
    v = (v > 0.f) ? v : NEG_SLOPE * v;
    float mm = m[d];
    if (!(mm > -INFINITY)) mm = 0.f;
    float p = expf(v - mm);
    p2[e] = p;
    atomicAdd(&dn[d], p);
}

constexpr int EPB2 = 16;
__global__ __launch_bounds__(64) void edge2_agg(
    const int* __restrict__ src, const int* __restrict__ dst,
    const float* __restrict__ p2, const float* __restrict__ dn,
    const float* __restrict__ h2, float* __restrict__ out2)
{
    const int t = threadIdx.x;
    const int e0 = blockIdx.x * EPB2;
#pragma unroll
    for (int i = 0; i < EPB2; ++i) {
        const int e = e0 + i;
        if (e >= ETOT) break;
        int s = (e < E_EDGES) ? src[e] : (e - E_EDGES);
        int d = (e < E_EDGES) ? dst[e] : (e - E_EDGES);
        if (t < NCLS) {
            float alpha = p2[e] / (dn[d] + 1e-16f);
            atomicAdd(&out2[(size_t)d * NCLS + t], alpha * h2[(size_t)s * NCLS + t]);
        }
    }
}

// ---------------- finalize: bias + log_softmax -> d_out (final, then logp) ----------------
__global__ void finalize(const float* __restrict__ out2, const float* __restrict__ b2,
                         float* __restrict__ out)
{
    int n = blockIdx.x * blockDim.x + threadIdx.x;
    if (n >= N_NODES) return;
    float v[NCLS];
    float mx = -INFINITY;
#pragma unroll
    for (int c = 0; c < NCLS; ++c) {
        v[c] = out2[(size_t)n * NCLS + c] + b2[c];
        mx = fmaxf(mx, v[c]);
        out[(size_t)n * NCLS + c] = v[c];
    }
    float s = 0.f;
#pragma unroll
    for (int c = 0; c < NCLS; ++c) s += expf(v[c] - mx);
    const float lse = mx + logf(s);
#pragma unroll
    for (int c = 0; c < NCLS; ++c)
        out[(size_t)N_NODES * NCLS + (size_t)n * NCLS + c] = v[c] - lse;
}

// ---------------- host launcher ----------------
extern "C" void kernel_launch(void* const* d_in, const int* in_sizes, int n_in,
                              void* d_out, int out_size, void* d_ws, size_t ws_size,
                              hipStream_t stream)
{
    (void)in_sizes; (void)n_in; (void)out_size; (void)ws_size;
    const float* x        = (const float*)d_in[0];
    const int*   ei       = (const int*)d_in[1];            // [2, E] int32
    // d_in[2] = edge_attr (unused by forward)
    const float* W1       = (const float*)d_in[3];
    const float* att_src1 = (const float*)d_in[4];
    const float* att_dst1 = (const float*)d_in[5];
    const float* b1       = (const float*)d_in[6];
    const float* W2       = (const float*)d_in[7];
    const float* att_src2 = (const float*)d_in[8];
    const float* att_dst2 = (const float*)d_in[9];
    const float* b2       = (const float*)d_in[10];
    float* out = (float*)d_out;

    const int* src = ei;
    const int* dst = ei + E_EDGES;

    // workspace carve-up (floats)
    float* ws = (float*)d_ws;
    size_t off = 0;
    float* h1   = ws + off; off += (size_t)N_NODES * HID;      // also reused for elu(h)
    float* aS1  = ws + off; off += (size_t)N_NODES * HEADS;
    float* aD1  = ws + off; off += (size_t)N_NODES * HEADS;
    float* m1   = ws + off; off += (size_t)N_NODES * HEADS;
    float* dn1  = ws + off; off += (size_t)N_NODES * HEADS;
    float* out1 = ws + off; off += (size_t)N_NODES * HID;
    float* p1   = ws + off; off += (size_t)ETOT * HEADS;
    float* h2   = ws + off; off += (size_t)N_NODES * NCLS;
    float* aS2  = ws + off; off += (size_t)N_NODES;
    float* aD2  = ws + off; off += (size_t)N_NODES;
    float* m2   = ws + off; off += (size_t)N_NODES;
    float* dn2  = ws + off; off += (size_t)N_NODES;
    float* p2   = ws + off; off += (size_t)ETOT;
    float* out2 = ws + off; off += (size_t)N_NODES * NCLS;

    // ---- layer 1 ----
    gemm1_wmma<<<N_NODES / 16, 256, 0, stream>>>(x, W1, h1);
    attn1_node<<<(N_NODES * HEADS + 255) / 256, 256, 0, stream>>>(
        h1, att_src1, att_dst1, aS1, aD1, m1, dn1, out1);
    edge1_max<<<(ETOT + 255) / 256, 256, 0, stream>>>(src, dst, aS1, aD1, m1);
    edge1_sum<<<(ETOT + 255) / 256, 256, 0, stream>>>(src, dst, aS1, aD1, m1, p1, dn1);
    edge1_agg<<<(ETOT + EPB1 - 1) / EPB1, 128, 0, stream>>>(src, dst, p1, dn1, h1, out1);
    node_mid<<<(N_NODES * HID + 255) / 256, 256, 0, stream>>>(out1, b1, h1); // h1 <- elu(out1+b1)

    // ---- layer 2 ----
    gemm2_wmma<<<N_NODES / 16, 96, 0, stream>>>(h1, W2, h2);
    attn2_node<<<(N_NODES + 255) / 256, 256, 0, stream>>>(
        h2, att_src2, att_dst2, aS2, aD2, m2, dn2, out2);
    edge2_max<<<(ETOT + 255) / 256, 256, 0, stream>>>(src, dst, aS2, aD2, m2);
    edge2_sum<<<(ETOT + 255) / 256, 256, 0, stream>>>(src, dst, aS2, aD2, m2, p2, dn2);
    edge2_agg<<<(ETOT + EPB2 - 1) / EPB2, 64, 0, stream>>>(src, dst, p2, dn2, h2, out2);
    finalize<<<(N_NODES + 255) / 256, 256, 0, stream>>>(out2, b2, out);
}